// SpatialSelfAttention_3839700763042
// MI455X (gfx1250) — compile-verified
//
#include <hip/hip_runtime.h>
#include <math.h>

// ---------------------------------------------------------------------------
// Problem constants (from reference)
// ---------------------------------------------------------------------------
#define BB   4
#define PP   12
#define NN   207
#define DD   64
#define HH_  4
#define DKK  16
#define DHH  32
#define TOK  (BB * PP * NN)          // 9936 tokens
#define JDIM (3 * HH_ * DKK)         // 192 = QKV "columns"
#define NEGF (-1.0e30f)

typedef float v2f __attribute__((ext_vector_type(2)));
typedef float v8f __attribute__((ext_vector_type(8)));

// D = A(16x4 f32) * B(4x16 f32) + C(16x16 f32)   -- fp32-exact matrix op
__device__ __forceinline__ v8f wmma_f32(v2f a, v2f b, v8f c) {
  return __builtin_amdgcn_wmma_f32_16x16x4_f32(
      /*neg_a=*/false, a, /*neg_b=*/false, b,
      /*c_mod=*/(short)0, c, /*reuse_a=*/false, /*reuse_b=*/false);
}

// Async DMA: 16 bytes per lane, global -> LDS, tracked by ASYNCcnt.
// Generic pointers to LDS carry the LDS byte offset in their low 32 bits.
__device__ __forceinline__ void async_g2l_b128(void* lds_ptr, const void* gptr) {
  unsigned lds = (unsigned)(uintptr_t)lds_ptr;
  asm volatile("global_load_async_to_lds_b128 %0, %1, off"
               :: "v"(lds), "v"(gptr) : "memory");
}
__device__ __forceinline__ void wait_async0() {
  asm volatile("s_wait_asynccnt 0x0" ::: "memory");
}

// ---------------------------------------------------------------------------
// Kernel 1: h[t,c] = relu( sum_d c_inputs[t,d] * W1[c,d] + b1[c] )
// one token per thread; W1 (32x64 = 8KB) in LDS, broadcast reads.
// ---------------------------------------------------------------------------
__global__ __launch_bounds__(256) void hidden_kernel(
    const float* __restrict__ cx, const float* __restrict__ W1,
    const float* __restrict__ b1, float* __restrict__ h) {
  __shared__ float W1s[DHH][DD];
  __shared__ float b1s[DHH];
  for (int i = threadIdx.x; i < DHH * DD; i += 256)
    W1s[i >> 6][i & 63] = W1[i];
  if (threadIdx.x < DHH) b1s[threadIdx.x] = b1[threadIdx.x];
  __syncthreads();

  const int t = blockIdx.x * 256 + threadIdx.x;
  if (t >= TOK) return;

  float row[DD];
#pragma unroll
  for (int d = 0; d < DD; ++d) row[d] = cx[t * DD + d];

  for (int c = 0; c < DHH; ++c) {
    float s = b1s[c];
#pragma unroll
    for (int d = 0; d < DD; ++d) s += row[d] * W1s[c][d];
    h[t * DHH + c] = fmaxf(s, 0.0f);
  }
}

// ---------------------------------------------------------------------------
// Kernel 2: the big bilinear GEMM (WMMA).
//   QKV[t, j] = sum_{c,d} h[t,c] * x[t,d] * W2[(j*64+d)*32 + c]
//            + sum_d     x[t,d] * b2[j*64+d]
// M = 64 tokens / block (4 WMMA tiles), N = 192 j (6 waves x 2 tiles),
// K = (d:64) x (c:32 + bias) ordered c-fastest so W2 rows stream contiguously.
// A fragment = rank-1 outer product built on the fly from LDS tiles.
// Tile loads use GLOBAL_LOAD_ASYNC_TO_LDS_B128 (ASYNCcnt-tracked DMA).
// ---------------------------------------------------------------------------
__global__ __launch_bounds__(192) void qkv_gemm_kernel(
    const float* __restrict__ x, const float* __restrict__ h,
    const float* __restrict__ W2, const float* __restrict__ b2,
    float* __restrict__ qkv) {
  __shared__ __align__(16) float hs[64][36];  // row stride 144 B (16B multiple)
  __shared__ __align__(16) float xs[64][68];  // row stride 272 B (16B multiple)

  const int tid = threadIdx.x;
  const int tbase = blockIdx.x * 64;

  // hs tile: 64 rows x 32 floats = 512 B128 chunks / 8-per-row
  for (int i = tid; i < 64 * 8; i += 192) {
    const int m = i >> 3, cq = (i & 7) << 2;
    const int t = tbase + m;
    if (t < TOK) {
      async_g2l_b128(&hs[m][cq], h + (size_t)t * DHH + cq);
    } else {
      float4 z = {0.f, 0.f, 0.f, 0.f};
      *(float4*)&hs[m][cq] = z;
    }
  }
  // xs tile: 64 rows x 64 floats, 16 B128 chunks per row
  for (int i = tid; i < 64 * 16; i += 192) {
    const int m = i >> 4, dq = (i & 15) << 2;
    const int t = tbase + m;
    if (t < TOK) {
      async_g2l_b128(&xs[m][dq], x + (size_t)t * DD + dq);
    } else {
      float4 z = {0.f, 0.f, 0.f, 0.f};
      *(float4*)&xs[m][dq] = z;
    }
  }
  wait_async0();
  __syncthreads();

  const int lane = tid & 31;
  const int wave = tid >> 5;      // 0..5
  const int half = lane >> 4;     // 0 | 1
  const int ln   = lane & 15;
  const int koff = half * 2;      // k pair {0,1} or {2,3}
  const int j0 = wave * 32;       // first j-tile
  const int j1 = wave * 32 + 16;  // second j-tile
  const int ja = j0 + ln;         // this lane's B column (tile 0)
  const int jb = j1 + ln;         // this lane's B column (tile 1)

  v8f acc[2][4];
#pragma unroll
  for (int i = 0; i < 2; ++i)
#pragma unroll
    for (int tt = 0; tt < 4; ++tt) acc[i][tt] = (v8f)0.0f;

  for (int d = 0; d < DD; ++d) {
    float xv[4];
#pragma unroll
    for (int tt = 0; tt < 4; ++tt) xv[tt] = xs[tt * 16 + ln][d];

    const float* w2ra = W2 + (ja * 64 + d) * 32;  // 32 contiguous c-values
    const float* w2rb = W2 + (jb * 64 + d) * 32;
    if (d + 1 < DD) {  // keep the W2 stream ahead of the matrix pipe
      __builtin_prefetch(W2 + (ja * 64 + d + 1) * 32, 0, 3);
      __builtin_prefetch(W2 + (jb * 64 + d + 1) * 32, 0, 3);
    }

#pragma unroll
    for (int cb = 0; cb < 32; cb += 4) {
      const int c = cb + koff;
      v2f bA; bA.x = w2ra[c]; bA.y = w2ra[c + 1];
      v2f bB; bB.x = w2rb[c]; bB.y = w2rb[c + 1];
#pragma unroll
      for (int tt = 0; tt < 4; ++tt) {
        v2f a;
        a.x = hs[tt * 16 + ln][c] * xv[tt];
        a.y = hs[tt * 16 + ln][c + 1] * xv[tt];
        acc[0][tt] = wmma_f32(a, bA, acc[0][tt]);
        acc[1][tt] = wmma_f32(a, bB, acc[1][tt]);
      }
    }
    // bias K-chunk: virtual c==32 row where h==1 and W2 row == b2
    {
      v2f bA; bA.x = half ? 0.0f : b2[ja * 64 + d]; bA.y = 0.0f;
      v2f bB; bB.x = half ? 0.0f : b2[jb * 64 + d]; bB.y = 0.0f;
#pragma unroll
      for (int tt = 0; tt < 4; ++tt) {
        v2f a; a.x = half ? 0.0f : xv[tt]; a.y = 0.0f;
        acc[0][tt] = wmma_f32(a, bA, acc[0][tt]);
        acc[1][tt] = wmma_f32(a, bB, acc[1][tt]);
      }
    }
  }

  // C/D layout: VGPR r holds row (r + 8*half), col = ln
#pragma unroll
  for (int tt = 0; tt < 4; ++tt) {
#pragma unroll
    for (int r = 0; r < 8; ++r) {
      const int t = tbase + tt * 16 + r + 8 * half;
      if (t < TOK) {
        qkv[t * JDIM + j0 + ln] = acc[0][tt][r];
        qkv[t * JDIM + j1 + ln] = acc[1][tt][r];
      }
    }
  }
}

// ---------------------------------------------------------------------------
// Kernel 3: masked-softmax attention per (b*p, head).  Block = 13 waves,
// wave = one 16-row tile.  scores and attn@V both via WMMA f32 16x16x4.
// QKV row layout per token: [Q(0..63) | K(64..127) | V(128..191)], head*16.
// ---------------------------------------------------------------------------
__global__ __launch_bounds__(416) void attention_kernel(
    const float* __restrict__ qkv, const float* __restrict__ tm,
    float* __restrict__ aout) {
  __shared__ float stg[13][16][20];  // per-wave 16x16 transpose staging

  const int tid  = threadIdx.x;
  const int lane = tid & 31;
  const int nt   = tid >> 5;        // wave id == row tile, 0..12
  const int bp   = blockIdx.x >> 2; // 0..47
  const int hh   = blockIdx.x & 3;  // head
  const int half = lane >> 4;
  const int ln   = lane & 15;
  const int koff = half * 2;
  const int tq_base = bp * NN;

  // ---- Q fragments (A matrix: 16 rows x 16 k, 4 chunks of K=4) ----
  v2f qa[4];
  {
    const int n = nt * 16 + ln;
    const bool v = (n < NN);
    const float* qrow = qkv + (tq_base + (v ? n : 0)) * JDIM + hh * DKK;
#pragma unroll
    for (int kc = 0; kc < 4; ++kc) {
      const int k = kc * 4 + koff;
      qa[kc].x = v ? qrow[k] : 0.0f;
      qa[kc].y = v ? qrow[k + 1] : 0.0f;
    }
  }

  // ---- scores = Q * K^T : 13 column tiles ----
  v8f acc[13];
#pragma unroll
  for (int mt = 0; mt < 13; ++mt) acc[mt] = (v8f)0.0f;

#pragma unroll
  for (int mt = 0; mt < 13; ++mt) {
    const int mc = mt * 16 + ln;
    const bool vm = (mc < NN);
    const float* krow = qkv + (tq_base + (vm ? mc : 0)) * JDIM + 64 + hh * DKK;
#pragma unroll
    for (int kc = 0; kc < 4; ++kc) {
      const int k = kc * 4 + koff;
      v2f kb;
      kb.x = vm ? krow[k] : 0.0f;
      kb.y = vm ? krow[k + 1] : 0.0f;
      acc[mt] = wmma_f32(qa[kc], kb, acc[mt]);
    }
  }

  // ---- mask + scale (pass 1) ----
#pragma unroll
  for (int mt = 0; mt < 13; ++mt) {
    const int mc = mt * 16 + ln;
#pragma unroll
    for (int r = 0; r < 8; ++r) {
      const int n = nt * 16 + r + 8 * half;
      const float tmv = (n < NN && mc < NN) ? tm[n * NN + mc] : 0.0f;
      const float s = acc[mt][r];
      acc[mt][r] = (tmv != 0.0f) ? s * 0.25f : NEGF;  // 1/sqrt(DK)=0.25
    }
  }

  // ---- row max (half-wave xor reduction keeps rows separate) ----
  float mx[8], sm[8];
#pragma unroll
  for (int r = 0; r < 8; ++r) {
    float v = acc[0][r];
#pragma unroll
    for (int mt = 1; mt < 13; ++mt) v = fmaxf(v, acc[mt][r]);
    for (int m = 1; m <= 8; m <<= 1) v = fmaxf(v, __shfl_xor(v, m, 32));
    mx[r] = v;
  }
  // ---- exp + row sum (pass 2) ----
#pragma unroll
  for (int mt = 0; mt < 13; ++mt)
#pragma unroll
    for (int r = 0; r < 8; ++r)
      acc[mt][r] = __expf(acc[mt][r] - mx[r]);
#pragma unroll
  for (int r = 0; r < 8; ++r) {
    float v = 0.0f;
#pragma unroll
    for (int mt = 0; mt < 13; ++mt) v += acc[mt][r];
    for (int m = 1; m <= 8; m <<= 1) v += __shfl_xor(v, m, 32);
    sm[r] = 1.0f / v;
  }

  // ---- attn = alpha * tm  ->  out = attn @ V (per tile, via LDS transpose)
  v8f acco = (v8f)0.0f;
  float (*st)[20] = stg[nt];
#pragma unroll
  for (int mt = 0; mt < 13; ++mt) {
    const int mc = mt * 16 + ln;
#pragma unroll
    for (int r = 0; r < 8; ++r) {
      const int n = nt * 16 + r + 8 * half;
      const float tmv = (n < NN && mc < NN) ? tm[n * NN + mc] : 0.0f;
      st[r + 8 * half][ln] = acc[mt][r] * tmv * sm[r];
    }
    // A = attn tile (rows = out rows, K = key index), B = V tile
#pragma unroll
    for (int kc = 0; kc < 4; ++kc) {
      const int krow = kc * 4 + koff;       // local key index
      v2f a; a.x = st[ln][krow]; a.y = st[ln][krow + 1];
      const int mg0 = mt * 16 + krow;
      v2f b;
      b.x = (mg0 < NN)     ? qkv[(tq_base + mg0) * JDIM + 128 + hh * DKK + ln] : 0.0f;
      b.y = (mg0 + 1 < NN) ? qkv[(tq_base + mg0 + 1) * JDIM + 128 + hh * DKK + ln] : 0.0f;
      acco = wmma_f32(a, b, acco);
    }
  }

#pragma unroll
  for (int r = 0; r < 8; ++r) {
    const int n = nt * 16 + r + 8 * half;
    if (n < NN) aout[(tq_base + n) * DD + hh * DKK + ln] = acco[r];
  }
}

// ---------------------------------------------------------------------------
// Kernel 4: out = aout @ Wf^T + inputs, then LayerNorm(gamma,beta).
// one token per thread; Wf (16KB) in LDS broadcast; two-pass LN via d_out.
// ---------------------------------------------------------------------------
__global__ __launch_bounds__(256) void final_kernel(
    const float* __restrict__ aout, const float* __restrict__ x,
    const float* __restrict__ Wf, const float* __restrict__ gamma,
    const float* __restrict__ beta, float* __restrict__ out) {
  __shared__ float WfS[DD][DD];
  for (int i = threadIdx.x; i < DD * DD; i += 256)
    WfS[i >> 6][i & 63] = Wf[i];
  __syncthreads();

  const int t = blockIdx.x * 256 + threadIdx.x;
  if (t >= TOK) return;

  float a[DD];
#pragma unroll
  for (int d = 0; d < DD; ++d) a[d] = aout[t * DD + d];

  float sum = 0.0f, sq = 0.0f;
  for (int o = 0; o < DD; ++o) {
    float s = 0.0f;
#pragma unroll
    for (int d = 0; d < DD; ++d) s += a[d] * WfS[o][d];
    const float y = s + x[t * DD + o];
    sum += y; sq += y * y;
    out[t * DD + o] = y;
  }
  const float mu  = sum * (1.0f / 64.0f);
  const float var = sq * (1.0f / 64.0f) - mu * mu;
  const float inv = rsqrtf(var + 1e-5f);
  for (int o = 0; o < DD; ++o) {
    const float y = out[t * DD + o];
    out[t * DD + o] = (y - mu) * inv * gamma[o] + beta[o];
  }
}

// ---------------------------------------------------------------------------
// Launch
// ---------------------------------------------------------------------------
extern "C" void kernel_launch(void* const* d_in, const int* in_sizes, int n_in,
                              void* d_out, int out_size, void* d_ws, size_t ws_size,
                              hipStream_t stream) {
  const float* inputs  = (const float*)d_in[0];
  const float* c_input = (const float*)d_in[1];
  const float* tmat    = (const float*)d_in[2];  // use transition_matrices[0]
  const float* W1      = (const float*)d_in[3];
  const float* b1      = (const float*)d_in[4];
  const float* W2      = (const float*)d_in[5];
  const float* b2      = (const float*)d_in[6];
  const float* Wf      = (const float*)d_in[7];
  const float* gamma   = (const float*)d_in[8];
  const float* beta    = (const float*)d_in[9];
  float* out = (float*)d_out;

  // workspace layout: h [TOK,32] | qkv [TOK,192] | aout [TOK,64]  (~11.5 MB)
  float* h    = (float*)d_ws;
  float* qkv  = h + (size_t)TOK * DHH;
  float* aout = qkv + (size_t)TOK * JDIM;

  hidden_kernel<<<(TOK + 255) / 256, 256, 0, stream>>>(c_input, W1, b1, h);
  qkv_gemm_kernel<<<(TOK + 63) / 64, 192, 0, stream>>>(inputs, h, W2, b2, qkv);
  attention_kernel<<<BB * PP * HH_, 13 * 32, 0, stream>>>(qkv, tmat, aout);
  final_kernel<<<(TOK + 255) / 256, 256, 0, stream>>>(aout, inputs, Wf, gamma, beta, out);
}